// SentenceTransformer_3255585211075
// MI455X (gfx1250) — compile-verified
//
#include <hip/hip_runtime.h>
#include <hip/hip_bf16.h>

// ---------------- problem constants ----------------
#define BQ      64          // queries
#define DDIM    256         // embedding dim
#define KTOP    16          // top-k
#define NBLK    256         // score-kernel grid
#define CHUNK   32          // memory rows staged per pass (2 tiles of 16)
#define LSTR    260         // padded LDS row stride (floats) -> conflict-free
#define MINSIM  0.1f

typedef float v2f __attribute__((ext_vector_type(2)));
typedef float v8f __attribute__((ext_vector_type(8)));

// ---------------------------------------------------------------------------
// Kernel 1: L2-normalize queries into workspace
// ---------------------------------------------------------------------------
__global__ __launch_bounds__(256) void normq_kernel(const float* __restrict__ Q,
                                                    float* __restrict__ Qn) {
  __shared__ float red[256];
  const int b = blockIdx.x, d = threadIdx.x;
  float v = Q[b * DDIM + d];
  red[d] = v * v;
  __syncthreads();
  for (int s = 128; s > 0; s >>= 1) {
    if (d < s) red[d] += red[d + s];
    __syncthreads();
  }
  const float nrm = sqrtf(red[0]);
  const float rin = nrm > 1e-12f ? 1.0f / nrm : 0.0f;
  Qn[b * DDIM + d] = v * rin;
}

// ---------------------------------------------------------------------------
// Kernel 2: stream M via async-to-LDS double buffering, WMMA fp32 sims,
//           per-block top-16 per query
// ---------------------------------------------------------------------------
__global__ __launch_bounds__(256) void score_kernel(const float* __restrict__ M,
                                                    const float* __restrict__ Qn,
                                                    float* __restrict__ candS,
                                                    int* __restrict__ candI,
                                                    int N) {
  __shared__ float stage[2][CHUNK * LSTR];  // 2 x 8320 floats (double buffer)
  __shared__ float sims[BQ * 36];           // 64 queries x 32 cols (padded)
  __shared__ float rinv[CHUNK];

  const int tid   = threadIdx.x;
  const int lane  = tid & 31;
  const int wave  = tid >> 5;
  const int qTile = wave & 3;   // which 16 queries
  const int nTile = wave >> 2;  // which 16 rows of the 32-row chunk
  const int col   = lane & 15;  // N column within tile / M row within A tile
  const int hi    = lane >> 4;  // 0: K pair {0,1}, 1: K pair {2,3}
  const int koff  = hi * 2;

  // --- per-thread staging geometry: 8 x b128 slots per chunk ---
  int srow[8], scol[8];
  #pragma unroll
  for (int t = 0; t < 8; ++t) {
    const int fl = t * 256 + tid;  // 0..2047 float4 slots
    srow[t] = fl >> 6;             // row 0..31
    scol[t] = (fl & 63) * 4;       // col 0..252
  }

  // --- preload A operand: 16x4 fp32 tiles over D=256 -> 64 v2f per lane ---
  v2f a[64];
  {
    const int q = qTile * 16 + col;
    const v2f* qp = (const v2f*)(Qn + q * DDIM + koff);
    #pragma unroll
    for (int s = 0; s < 64; ++s) a[s] = qp[2 * s];  // elems {4s+koff, 4s+koff+1}
  }

  const int perB = (N + NBLK - 1) / NBLK;
  const int slab = ((perB + CHUNK - 1) / CHUNK) * CHUNK;
  const int base = blockIdx.x * slab;

  // issue 8 async global->LDS b128 loads for one chunk (GVS addressing)
  auto issue = [&](int rowBase, int buf) {
    const float* sbase = M + (size_t)rowBase * DDIM;  // uniform -> SGPR pair
    #pragma unroll
    for (int t = 0; t < 8; ++t) {
      int gr = rowBase + srow[t];
      if (gr > N - 1) gr = N - 1;                     // clamp (rinv kills OOB rows)
      const int voff = (gr - rowBase) * (DDIM * 4) + scol[t] * 4;  // signed i32
      const unsigned lds = (unsigned)(size_t)&stage[buf][srow[t] * LSTR + scol[t]];
      asm volatile("global_load_async_to_lds_b128 %0, %1, %2"
                   :: "v"(lds), "v"(voff), "s"(sbase)
                   : "memory");
    }
  };

  // thread-local top-16 (kept in VGPRs: only constant-index accesses below)
  float ts[KTOP];
  int   ti[KTOP];
  #pragma unroll
  for (int j = 0; j < KTOP; ++j) { ts[j] = -1e30f; ti[j] = 0x7FFFFFFF; }
  float tmin = -1e30f;

  issue(base, 0);  // prologue: chunk 0 -> buffer 0

  for (int c0 = 0; c0 < slab; c0 += CHUNK) {
    const int rowBase = base + c0;
    const int buf = (c0 / CHUNK) & 1;

    // overlap: kick off next chunk, then wait only for the current one.
    // Async loads complete in order, so <=8 outstanding => current chunk done.
    if (c0 + CHUNK < slab) {
      issue(rowBase + CHUNK, buf ^ 1);
      asm volatile("s_wait_asynccnt 0x8" ::: "memory");
    } else {
      asm volatile("s_wait_asynccnt 0x0" ::: "memory");
    }
    __syncthreads();

    const float* sb = &stage[buf][0];

    // ---- per-row inverse norms ----
    if (tid < CHUNK) {
      float ss = 0.f;
      const float* rp = &sb[tid * LSTR];
      #pragma unroll 8
      for (int d = 0; d < DDIM; ++d) ss += rp[d] * rp[d];
      const float nrm = sqrtf(ss);
      rinv[tid] = (nrm > 1e-12f && (rowBase + tid) < N) ? (1.0f / nrm) : 0.0f;
    }
    __syncthreads();

    // ---- 16x16 sims tile via 64 x V_WMMA_F32_16X16X4_F32 ----
    v8f acc = {0.f, 0.f, 0.f, 0.f, 0.f, 0.f, 0.f, 0.f};
    const v2f* bp = (const v2f*)(&sb[(nTile * 16 + col) * LSTR + koff]);
    #pragma unroll
    for (int s = 0; s < 64; ++s) {
      const v2f b = bp[2 * s];  // B[k][n]: row n elems {4s+koff, 4s+koff+1}
      acc = __builtin_amdgcn_wmma_f32_16x16x4_f32(false, a[s], false, b,
                                                  (short)0, acc, false, false);
    }
    const float rin = rinv[nTile * 16 + col];
    #pragma unroll
    for (int r = 0; r < 8; ++r) {
      const int q = qTile * 16 + r + hi * 8;  // C/D layout: vgpr r -> M=r / r+8
      sims[q * 36 + nTile * 16 + col] = acc[r] * rin;
    }
    __syncthreads();

    // ---- top-k scan: 4 threads per query, 8 columns each ----
    {
      const int q = tid >> 2, sub = tid & 3;
      #pragma unroll
      for (int j = 0; j < 8; ++j) {
        const int n = sub + j * 4;
        const float v = sims[q * 36 + n];
        if (v > tmin) {
          int mp = 0; float mv = ts[0];
          #pragma unroll
          for (int k = 1; k < KTOP; ++k)
            if (ts[k] < mv) { mv = ts[k]; mp = k; }
          #pragma unroll
          for (int k = 0; k < KTOP; ++k)
            if (k == mp) { ts[k] = v; ti[k] = rowBase + n; }
          tmin = ts[0];
          #pragma unroll
          for (int k = 1; k < KTOP; ++k) tmin = fminf(tmin, ts[k]);
        }
      }
    }
    __syncthreads();  // also fences buffer reuse for the chunk issued above
  }

  // ---- merge 4 thread-lists per query -> block top-16, write candidates ----
  float* mS = &stage[0][0];                 // 4096 floats
  int*   mI = (int*)(&stage[0][0] + 4096);  // 4096 ints (buffer has 8320 floats)
  {
    const int q = tid >> 2, sub = tid & 3;
    #pragma unroll
    for (int j = 0; j < KTOP; ++j) {
      mS[q * 64 + sub * 16 + j] = ts[j];
      mI[q * 64 + sub * 16 + j] = ti[j];
    }
  }
  __syncthreads();
  if (tid < BQ) {
    const int q = tid;
    float* s = &mS[q * 64];
    int*   ii = &mI[q * 64];
    for (int o = 0; o < KTOP; ++o) {
      int bpi = 0; float bv = s[0]; int bi = ii[0];
      for (int j = 1; j < 64; ++j) {
        const float v = s[j];
        if (v > bv || (v == bv && ii[j] < bi)) { bv = v; bi = ii[j]; bpi = j; }
      }
      candS[((size_t)blockIdx.x * BQ + q) * KTOP + o] = bv;
      candI[((size_t)blockIdx.x * BQ + q) * KTOP + o] = bi;
      s[bpi] = -1e30f;
    }
  }
}

// ---------------------------------------------------------------------------
// Kernel 3: merge per-block candidates -> global top-16, threshold, compact
// ---------------------------------------------------------------------------
__global__ __launch_bounds__(256) void reduce_kernel(const float* __restrict__ candS,
                                                     const int* __restrict__ candI,
                                                     int* __restrict__ finI,
                                                     int* __restrict__ finK,
                                                     float* __restrict__ outMask,
                                                     int nb) {
  __shared__ float rv[256];
  __shared__ int   rid[256];
  __shared__ int   rt[256];
  __shared__ float selS[KTOP];
  __shared__ int   selI[KTOP];

  const int b = blockIdx.x;   // query
  const int tid = threadIdx.x;
  const int total = nb * KTOP;

  float ts[KTOP]; int ti[KTOP];
  #pragma unroll
  for (int j = 0; j < KTOP; ++j) { ts[j] = -1e30f; ti[j] = 0x7FFFFFFF; }

  for (int c = tid; c < total; c += 256) {
    const int blk = c >> 4, o = c & 15;
    const float v = candS[((size_t)blk * BQ + b) * KTOP + o];
    const int  id = candI[((size_t)blk * BQ + b) * KTOP + o];
    int wp = 0; float wv = ts[0]; int wi = ti[0];
    #pragma unroll
    for (int j = 1; j < KTOP; ++j)
      if (ts[j] < wv || (ts[j] == wv && ti[j] > wi)) { wv = ts[j]; wi = ti[j]; wp = j; }
    if (v > wv || (v == wv && id < wi)) {
      #pragma unroll
      for (int j = 0; j < KTOP; ++j)
        if (j == wp) { ts[j] = v; ti[j] = id; }
    }
  }

  // local sort descending (sim desc, idx asc), constant indices only
  #pragma unroll
  for (int i = 0; i < KTOP - 1; ++i) {
    int bp = i;
    #pragma unroll
    for (int j = 0; j < KTOP; ++j)
      if (j > i && (ts[j] > ts[bp] || (ts[j] == ts[bp] && ti[j] < ti[bp]))) bp = j;
    #pragma unroll
    for (int j = 0; j < KTOP; ++j)
      if (j == bp) {
        const float tv = ts[i]; const int tt = ti[i];
        ts[i] = ts[j]; ti[i] = ti[j];
        ts[j] = tv;    ti[j] = tt;
      }
  }

  // 16 rounds of block-wide argmax over the heads of 256 sorted lists
  for (int o = 0; o < KTOP; ++o) {
    rv[tid] = ts[0]; rid[tid] = ti[0]; rt[tid] = tid;
    __syncthreads();
    for (int s = 128; s > 0; s >>= 1) {
      if (tid < s) {
        const bool take = (rv[tid + s] > rv[tid]) ||
                          (rv[tid + s] == rv[tid] && rid[tid + s] < rid[tid]);
        if (take) { rv[tid] = rv[tid + s]; rid[tid] = rid[tid + s]; rt[tid] = rt[tid + s]; }
      }
      __syncthreads();
    }
    if (tid == 0) { selS[o] = rv[0]; selI[o] = rid[0]; }
    const int w = rt[0];
    __syncthreads();
    if (tid == w) {  // pop the winner's head
      #pragma unroll
      for (int j = 0; j < KTOP - 1; ++j) { ts[j] = ts[j + 1]; ti[j] = ti[j + 1]; }
      ts[KTOP - 1] = -1e30f; ti[KTOP - 1] = 0x7FFFFFFF;
    }
    __syncthreads();
  }

  if (tid == 0) {
    int nval = 0;
    int cidx[KTOP];
    for (int o = 0; o < KTOP; ++o)
      if (selS[o] >= MINSIM) cidx[nval++] = selI[o];
    const bool use = (nval >= 2);
    for (int o = 0; o < KTOP; ++o) {
      const int keep = (o < nval && use) ? 1 : 0;
      finI[b * KTOP + o] = (o < nval) ? cidx[o] : 0;
      finK[b * KTOP + o] = keep;
      outMask[b * KTOP + o] = keep ? 0.0f : 1.0f;  // True where padded/invalid
    }
  }
}

// ---------------------------------------------------------------------------
// Kernel 4: gather + re-normalize winning rows (zeros where masked)
// ---------------------------------------------------------------------------
__global__ __launch_bounds__(256) void gather_kernel(const float* __restrict__ M,
                                                     const int* __restrict__ finI,
                                                     const int* __restrict__ finK,
                                                     float* __restrict__ outE) {
  __shared__ float red[256];
  const int slot = blockIdx.x;   // b*16 + k
  const int d = threadIdx.x;
  const int keep = finK[slot];
  const int idx = finI[slot];
  float v = 0.f;
  if (keep) v = M[(size_t)idx * DDIM + d];
  red[d] = v * v;
  __syncthreads();
  for (int s = 128; s > 0; s >>= 1) {
    if (d < s) red[d] += red[d + s];
    __syncthreads();
  }
  const float nrm = sqrtf(red[0]);
  const float rin = nrm > 1e-12f ? 1.0f / nrm : 0.0f;
  outE[(size_t)slot * DDIM + d] = keep ? v * rin : 0.0f;
}

// ---------------------------------------------------------------------------
extern "C" void kernel_launch(void* const* d_in, const int* in_sizes, int n_in,
                              void* d_out, int out_size, void* d_ws, size_t ws_size,
                              hipStream_t stream) {
  const float* Q = (const float*)d_in[0];   // [64, 256]
  const float* M = (const float*)d_in[1];   // [N, 256]
  const int N = in_sizes[1] / DDIM;

  float* ws    = (float*)d_ws;
  float* Qn    = ws;                               // 64*256 floats
  float* candS = Qn + BQ * DDIM;                   // NBLK*64*16 floats
  int*   candI = (int*)(candS + (size_t)NBLK * BQ * KTOP);
  int*   finI  = candI + (size_t)NBLK * BQ * KTOP; // 64*16 ints
  int*   finK  = finI + BQ * KTOP;                 // 64*16 ints

  float* outE    = (float*)d_out;                  // [64,16,256]
  float* outMask = outE + (size_t)BQ * KTOP * DDIM;// [64,16]

  normq_kernel<<<BQ, 256, 0, stream>>>(Q, Qn);
  score_kernel<<<NBLK, 256, 0, stream>>>(M, Qn, candS, candI, N);
  reduce_kernel<<<BQ, 256, 0, stream>>>(candS, candI, finI, finK, outMask, NBLK);
  gather_kernel<<<BQ * KTOP, 256, 0, stream>>>(M, finI, finK, outE);
}